// TempAttention_66760971649601
// MI455X (gfx1250) — compile-verified
//
#include <hip/hip_runtime.h>

// ---------------------------------------------------------------------------
// TempAttention for MI455X (gfx1250, wave32)
//   Phase 1: two tiny fp32 attentions via V_WMMA_F32_16X16X4_F32 -> d_ws
//   Phase 2: bandwidth-bound broadcast-add (float4, non-temporal stores)
// ---------------------------------------------------------------------------

typedef float v2f __attribute__((ext_vector_type(2)));
typedef float v4f __attribute__((ext_vector_type(4)));
typedef float v8f __attribute__((ext_vector_type(8)));

#define NB 16   // batch
#define NH 8    // heads
#define NE 64   // head dim
#define PATCH_NUM  32
#define PATCH_SIZE 128
#define WIN_SIZE   4096

// One workgroup per (b,h). blockDim = (32, L/16): wave y owns query rows
// [16y, 16y+16). Full softmax attention, scale = 1/sqrt(E) = 0.125.
template <int L>
__global__ __launch_bounds__(32 * (L / 16))
void attn_wmma_kernel(const float* __restrict__ Q,
                      const float* __restrict__ K,
                      const float* __restrict__ V,
                      float* __restrict__ O)
{
    constexpr int NT = L / 16;                  // 16-wide column tiles
    __shared__ float p_lds[NT * 16 * L];        // per-wave P staging, row-major [16][L]

    const int bh   = blockIdx.x;
    const int b    = bh / NH;
    const int h    = bh % NH;
    const int wave = threadIdx.y;               // query row-block id
    const int qr0  = wave * 16;
    const int lane = threadIdx.x;
    const int lo   = lane & 15;                 // N / M-in-half index
    const int hi   = lane >> 4;                 // half-wave select
    const float scale = 0.125f;

    // elem(b, row, h, e) = ((b*L + row)*NH + h)*NE + e   (row-major [B,L,H,E])
    const size_t rowQ = ((size_t)b * L + qr0 + lo) * NH + h;   // this lane's Q row
    const size_t rowK = ((size_t)b * L +        lo) * NH + h;  // K row base (add tile*16 rows)

    // ---- S = scale * Q K^T : NT tiles of 16x16, K-loop over E in steps of 4 ----
    v8f s[NT];
#pragma unroll
    for (int t = 0; t < NT; ++t) {
        v8f acc = {};
#pragma unroll
        for (int k0 = 0; k0 < NE; k0 += 4) {
            v2f a  = *(const v2f*)(Q + rowQ * NE + k0 + 2 * hi);
            v2f bb = *(const v2f*)(K + (rowK + (size_t)t * 16 * NH) * NE + k0 + 2 * hi);
            acc = __builtin_amdgcn_wmma_f32_16x16x4_f32(
                      false, a, false, bb, (short)0, acc, false, false);
        }
        s[t] = acc * scale;
    }

    // ---- row softmax: row (v + 8*hi) lives across a 16-lane half-wave ----
#pragma unroll
    for (int v = 0; v < 8; ++v) {
        float m = -3.402823466e38f;
#pragma unroll
        for (int t = 0; t < NT; ++t) m = fmaxf(m, s[t][v]);
#pragma unroll
        for (int off = 1; off < 16; off <<= 1) m = fmaxf(m, __shfl_xor(m, off, 16));
        float sum = 0.f;
#pragma unroll
        for (int t = 0; t < NT; ++t) {
            float p = __expf(s[t][v] - m);
            s[t][v] = p;
            sum += p;
        }
#pragma unroll
        for (int off = 1; off < 16; off <<= 1) sum += __shfl_xor(sum, off, 16);
        float inv = 1.0f / sum;
#pragma unroll
        for (int t = 0; t < NT; ++t) s[t][v] *= inv;
    }

    // ---- stage P (C/D layout) -> LDS row-major [16][L], per-wave region ----
    float* pw = &p_lds[wave * 16 * L];
#pragma unroll
    for (int t = 0; t < NT; ++t)
#pragma unroll
        for (int v = 0; v < 8; ++v)
            pw[(v + 8 * hi) * L + t * 16 + lo] = s[t][v];
    __syncthreads();

    // ---- O = P V : 4 tiles of 16x16 over E, K-loop over L in steps of 4 ----
#pragma unroll
    for (int e0 = 0; e0 < NE; e0 += 16) {
        v8f acc = {};
#pragma unroll
        for (int k0 = 0; k0 < L; k0 += 4) {
            // A-layout reload of P: element [lo][k0 + 2*hi + v]
            v2f a = *(const v2f*)&pw[lo * L + k0 + 2 * hi];
            // B-layout V: element [k0 + 2*hi + v][e0 + lo]; rows strided by NH*NE
            v2f bb;
            bb.x = V[(((size_t)b * L + k0 + 2 * hi + 0) * NH + h) * NE + e0 + lo];
            bb.y = V[(((size_t)b * L + k0 + 2 * hi + 1) * NH + h) * NE + e0 + lo];
            acc = __builtin_amdgcn_wmma_f32_16x16x4_f32(
                      false, a, false, bb, (short)0, acc, false, false);
        }
#pragma unroll
        for (int v = 0; v < 8; ++v)
            O[(((size_t)b * L + qr0 + v + 8 * hi) * NH + h) * NE + e0 + lo] = acc[v];
    }
}

// out[b,i,h,e] = inter[b, i>>7, h, e] + intra[b, i>>5, h, e]; float4 per thread.
// Output is a 134 MB write-once stream -> non-temporal stores; the 5 MB of
// intermediates are re-read from L2 (192 MB).
__global__ __launch_bounds__(256)
void bcast_add_kernel(const float* __restrict__ inter,
                      const float* __restrict__ intra,
                      float* __restrict__ out)
{
    size_t idx = (size_t)blockIdx.x * blockDim.x + threadIdx.x;   // float4 index
    int e4 = (int)(idx & 15);          // E/4 = 16 float4 per (b,i,h)
    size_t t = idx >> 4;
    int h = (int)(t & 7);  t >>= 3;
    int i = (int)(t & 4095); t >>= 12;
    int b = (int)t;

    const v4f* inter4 = (const v4f*)inter;
    const v4f* intra4 = (const v4f*)intra;
    v4f a = inter4[(((size_t)b * PATCH_NUM  + (i >> 7)) * NH + h) * 16 + e4];
    v4f c = intra4[(((size_t)b * PATCH_SIZE + (i >> 5)) * NH + h) * 16 + e4];
    v4f r = a + c;
    __builtin_nontemporal_store(r, (v4f*)out + idx);
}

extern "C" void kernel_launch(void* const* d_in, const int* in_sizes, int n_in,
                              void* d_out, int out_size, void* d_ws, size_t ws_size,
                              hipStream_t stream)
{
    const float* q_inter = (const float*)d_in[0];
    const float* k_inter = (const float*)d_in[1];
    const float* v_inter = (const float*)d_in[2];
    const float* q_intra = (const float*)d_in[3];
    const float* k_intra = (const float*)d_in[4];
    const float* v_intra = (const float*)d_in[5];

    // workspace: V_inter (1 MB) then V_intra (4 MB)
    float* Vi = (float*)d_ws;
    float* Va = Vi + (size_t)NB * PATCH_NUM * NH * NE;

    attn_wmma_kernel<PATCH_NUM><<<dim3(NB * NH), dim3(32, PATCH_NUM / 16), 0, stream>>>(
        q_inter, k_inter, v_inter, Vi);
    attn_wmma_kernel<PATCH_SIZE><<<dim3(NB * NH), dim3(32, PATCH_SIZE / 16), 0, stream>>>(
        q_intra, k_intra, v_intra, Va);

    const size_t n4 = (size_t)NB * WIN_SIZE * NH * NE / 4;   // 8,388,608 float4
    bcast_add_kernel<<<(unsigned)(n4 / 256), 256, 0, stream>>>(Vi, Va, (float*)d_out);
}